// Summarizer_39522289058331
// MI455X (gfx1250) — compile-verified
//
#include <hip/hip_runtime.h>

typedef __attribute__((ext_vector_type(2))) float v2f;
typedef __attribute__((ext_vector_type(8))) float v8f;

namespace {
constexpr int Bc = 8, Tc = 20, Nc = 10000, Fc = 9, Ec = 160000, Hc = 64;
constexpr int BN  = Bc * Nc;        // 80000 GEMM rows
constexpr int TBN = Tc * BN;        // 1,600,000 s entries
constexpr int OUTF = Fc - 1 + Hc;   // 72 output features
}

__device__ __forceinline__ void fatomic_add(float* p, float v) {
  __hip_atomic_fetch_add(p, v, __ATOMIC_RELAXED, __HIP_MEMORY_SCOPE_AGENT);
}

__global__ void k_deg_init(float* __restrict__ deg) {
  int i = blockIdx.x * blockDim.x + threadIdx.x;
  if (i < Nc) deg[i] = 1.0f;  // self-loop contributes 1 to every node
}

__global__ void k_zero4(float4* __restrict__ p, int n4) {
  int i = blockIdx.x * blockDim.x + threadIdx.x;
  if (i < n4) p[i] = make_float4(0.f, 0.f, 0.f, 0.f);
}

__global__ void k_deg_accum(const long long* __restrict__ ei, float* __restrict__ deg) {
  int e = blockIdx.x * blockDim.x + threadIdx.x;
  if (e < Ec) fatomic_add(&deg[(int)ei[Ec + e]], 1.0f);
}

__global__ void k_dinv(float* __restrict__ deg) {
  int i = blockIdx.x * blockDim.x + threadIdx.x;
  if (i < Nc) deg[i] = 1.0f / sqrtf(deg[i]);  // deg >= 1 always (self-loops)
}

__global__ void k_norm(const long long* __restrict__ ei, const float* __restrict__ dinv,
                       float* __restrict__ nrm) {
  int e = blockIdx.x * blockDim.x + threadIdx.x;
  if (e < Ec) nrm[e] = dinv[(int)ei[e]] * dinv[(int)ei[Ec + e]];
}

// s[t*BN + b*N + n] = dinv[n]^2 * x[b,t,n]  (self-loop term)
__global__ void k_s_init(const float* __restrict__ obs, const float* __restrict__ dinv,
                         float* __restrict__ s) {
  int j = blockIdx.x * blockDim.x + threadIdx.x;
  if (j >= TBN) return;
  int t = j / BN, r = j - t * BN;
  int b = r / Nc, n = r - b * Nc;
  float d = dinv[n];
  s[j] = d * d * obs[((size_t)(b * Tc + t) * Nc + n) * Fc + (Fc - 1)];
}

// scatter: s[t,b,dst[e]] += norm[e] * x[b,t,src[e]]  for all (b,t,e)
__global__ void k_s_edges(const long long* __restrict__ ei, const float* __restrict__ nrm,
                          const float* __restrict__ obs, float* __restrict__ s) {
  long long j = (long long)blockIdx.x * blockDim.x + threadIdx.x;
  if (j >= (long long)Ec * Bc * Tc) return;
  int e  = (int)(j % Ec);
  int bt = (int)(j / Ec);
  int b = bt / Tc, t = bt - b * Tc;
  int srcn = (int)ei[e], dstn = (int)ei[Ec + e];
  float msg = nrm[e] * obs[((size_t)(b * Tc + t) * Nc + srcn) * Fc + (Fc - 1)];
  fatomic_add(&s[(size_t)t * BN + (size_t)b * Nc + dstn], msg);
}

// out[..., 0:8] = observations[..., 0:8]
__global__ void k_copy_other(const float* __restrict__ obs, float* __restrict__ out) {
  int j = blockIdx.x * blockDim.x + threadIdx.x;
  if (j >= Bc * Tc * Nc * (Fc - 1)) return;
  int btn = j / (Fc - 1), f = j - btn * (Fc - 1);
  out[(size_t)btn * OUTF + f] = obs[(size_t)btn * Fc + f];
}

// Build B_ext[gate][K=68][64] in WMMA-pair-interleaved form:
// Uext[((g*34 + k/2)*64 + n)*2 + (k&1)] ; rows 0..63 = Wl_g[H+k][n],
// row 64 = a_g[n] = Wc_g @ Wl_g[:H], row 65 = c_g[n] = bc_g @ Wl_g[:H] + bl_g, rows 66/67 = 0
__global__ void k_uext(const float* __restrict__ Wc_z, const float* __restrict__ bc_z,
                       const float* __restrict__ Wl_z, const float* __restrict__ bl_z,
                       const float* __restrict__ Wc_r, const float* __restrict__ bc_r,
                       const float* __restrict__ Wl_r, const float* __restrict__ bl_r,
                       const float* __restrict__ Wc_h, const float* __restrict__ bc_h,
                       const float* __restrict__ Wl_h, const float* __restrict__ bl_h,
                       float* __restrict__ Uext) {
  int j = blockIdx.x * blockDim.x + threadIdx.x;
  if (j >= 3 * 68 * 64) return;
  int g = j / (68 * 64), r = j % (68 * 64);
  int k = r / 64, n = r - k * 64;
  const float* Wc = g == 0 ? Wc_z : g == 1 ? Wc_r : Wc_h;
  const float* bc = g == 0 ? bc_z : g == 1 ? bc_r : bc_h;
  const float* Wl = g == 0 ? Wl_z : g == 1 ? Wl_r : Wl_h;
  const float* bl = g == 0 ? bl_z : g == 1 ? bl_r : bl_h;
  float v = 0.f;
  if (k < Hc) {
    v = Wl[(Hc + k) * Hc + n];
  } else if (k == Hc) {
    float a = 0.f;
    for (int i = 0; i < Hc; ++i) a += Wc[i] * Wl[i * Hc + n];
    v = a;
  } else if (k == Hc + 1) {
    float a = bl[n];
    for (int i = 0; i < Hc; ++i) a += bc[i] * Wl[i * Hc + n];
    v = a;
  }
  Uext[((g * 34 + (k >> 1)) * 64 + n) * 2 + (k & 1)] = v;
}

// One GRU timestep: per wave one 16-row tile of the (80000 x 64) state.
// pre_g = [Hst | s | 1] @ [U_g; a_g; c_g]  via v_wmma_f32_16x16x4_f32, K=68.
__global__ __launch_bounds__(256) void k_gru_step(
    const float* __restrict__ Hcur, float* __restrict__ Hnext,
    const float* __restrict__ s_t, const float* __restrict__ Uexf,
    float* __restrict__ out, int tstep) {
  __shared__ __align__(16) float Gt[8 * 16 * 66];  // per-wave Hst*R tile, stride 66 keeps b64 aligned
  const v2f* __restrict__ U2 = (const v2f*)Uexf;
  const int wave = threadIdx.x >> 5, lane = threadIdx.x & 31;
  const int half = lane >> 4, r16 = lane & 15;
  const int row0 = (blockIdx.x * 8 + wave) << 4;
  const v8f vzero = {};

  // A fragments: lane layout per ISA (lanes 0-15: K=4kk..4kk+1, lanes 16-31: K=4kk+2..4kk+3)
  v2f a[17];
  {
    const float* hr = Hcur + (size_t)(row0 + r16) * Hc + half * 2;
#pragma unroll
    for (int kk = 0; kk < 16; ++kk) a[kk] = *(const v2f*)(hr + kk * 4);
    float sv = s_t[row0 + r16];
    v2f ext; ext.x = half ? 0.f : sv; ext.y = half ? 0.f : 1.f;
    a[16] = ext;  // K=64 -> s, K=65 -> 1, K=66/67 -> 0
  }

  // Hst in C/D layout (VGPR q: M = q + 8*half, N = r16 + 16*t4)
  float hcd[4][8];
#pragma unroll
  for (int q = 0; q < 8; ++q) {
    const float* hr = Hcur + (size_t)(row0 + q + half * 8) * Hc + r16;
#pragma unroll
    for (int t4 = 0; t4 < 4; ++t4) hcd[t4][q] = hr[t4 * 16];
  }

  // z and r gates (interleave 8 accumulator chains for ILP)
  v8f accz[4], accr[4];
#pragma unroll
  for (int t4 = 0; t4 < 4; ++t4) { accz[t4] = vzero; accr[t4] = vzero; }
#pragma unroll
  for (int kk = 0; kk < 17; ++kk) {
    int p = (2 * kk + half) * 64 + r16;
#pragma unroll
    for (int t4 = 0; t4 < 4; ++t4) {
      v2f bz = U2[p + t4 * 16];
      v2f br = U2[34 * 64 + p + t4 * 16];
      accz[t4] = __builtin_amdgcn_wmma_f32_16x16x4_f32(false, a[kk], false, bz, (short)0, accz[t4], false, false);
      accr[t4] = __builtin_amdgcn_wmma_f32_16x16x4_f32(false, a[kk], false, br, (short)0, accr[t4], false, false);
    }
  }

  // Z = sigmoid(pre_z); G = Hst * sigmoid(pre_r) -> LDS (C/D layout -> row-major tile)
  float* gw = &Gt[wave * (16 * 66)];
#pragma unroll
  for (int t4 = 0; t4 < 4; ++t4) {
#pragma unroll
    for (int q = 0; q < 8; ++q) {
      float z = 1.f / (1.f + __expf(-accz[t4][q]));
      accz[t4][q] = z;
      float rr = 1.f / (1.f + __expf(-accr[t4][q]));
      gw[(q + half * 8) * 66 + r16 + t4 * 16] = hcd[t4][q] * rr;
    }
  }

  // Re-read G as A fragments (same-wave LDS ops are in-order; no barrier needed)
  v2f ga[17];
  {
    const float* gr = gw + r16 * 66 + half * 2;
#pragma unroll
    for (int kk = 0; kk < 16; ++kk) ga[kk] = *(const v2f*)(gr + kk * 4);
    ga[16] = a[16];
  }

  // h gate
  v8f acch[4];
#pragma unroll
  for (int t4 = 0; t4 < 4; ++t4) acch[t4] = vzero;
#pragma unroll
  for (int kk = 0; kk < 17; ++kk) {
    int p = 2 * 34 * 64 + (2 * kk + half) * 64 + r16;
#pragma unroll
    for (int t4 = 0; t4 < 4; ++t4) {
      v2f bh = U2[p + t4 * 16];
      acch[t4] = __builtin_amdgcn_wmma_f32_16x16x4_f32(false, ga[kk], false, bh, (short)0, acch[t4], false, false);
    }
  }

  // Hn = Z*Hst + (1-Z)*tanh(pre_h); write state + output slice
#pragma unroll
  for (int q = 0; q < 8; ++q) {
    int row = row0 + q + half * 8;
    int b = row / Nc, n = row - b * Nc;
    float* orow = out + ((size_t)(b * Tc + tstep) * Nc + n) * OUTF + (Fc - 1) + r16;
    float* hnr  = Hnext + (size_t)row * Hc + r16;
#pragma unroll
    for (int t4 = 0; t4 < 4; ++t4) {
      float z  = accz[t4][q];
      float ht = tanhf(acch[t4][q]);
      float hn = z * hcd[t4][q] + (1.f - z) * ht;
      hnr[t4 * 16]  = hn;
      orow[t4 * 16] = hn;
    }
  }
}

extern "C" void kernel_launch(void* const* d_in, const int* in_sizes, int n_in,
                              void* d_out, int out_size, void* d_ws, size_t ws_size,
                              hipStream_t stream) {
  const float*     obs = (const float*)d_in[0];
  const long long* ei  = (const long long*)d_in[1];
  const float *Wc_z = (const float*)d_in[2],  *bc_z = (const float*)d_in[3];
  const float *Wl_z = (const float*)d_in[4],  *bl_z = (const float*)d_in[5];
  const float *Wc_r = (const float*)d_in[6],  *bc_r = (const float*)d_in[7];
  const float *Wl_r = (const float*)d_in[8],  *bl_r = (const float*)d_in[9];
  const float *Wc_h = (const float*)d_in[10], *bc_h = (const float*)d_in[11];
  const float *Wl_h = (const float*)d_in[12], *bl_h = (const float*)d_in[13];
  float* out = (float*)d_out;
  char* ws = (char*)d_ws;

  size_t off = 0;
  float* Ha   = (float*)(ws + off); off += (size_t)BN * Hc * 4;   // 20.48 MB
  float* Hb   = (float*)(ws + off); off += (size_t)BN * Hc * 4;   // 20.48 MB
  float* s    = (float*)(ws + off); off += (size_t)TBN * 4;       // 6.4 MB
  float* deg  = (float*)(ws + off); off += (size_t)Nc * 4;        // deg -> dinv in place
  float* nrm  = (float*)(ws + off); off += (size_t)Ec * 4;
  float* Uext = (float*)(ws + off); off += (size_t)3 * 68 * 64 * 4;

  dim3 blk(256);
  k_deg_init<<<(Nc + 255) / 256, blk, 0, stream>>>(deg);
  k_zero4<<<(BN * Hc / 4 + 255) / 256, blk, 0, stream>>>((float4*)Ha, BN * Hc / 4);
  k_deg_accum<<<(Ec + 255) / 256, blk, 0, stream>>>(ei, deg);
  k_dinv<<<(Nc + 255) / 256, blk, 0, stream>>>(deg);
  k_norm<<<(Ec + 255) / 256, blk, 0, stream>>>(ei, deg, nrm);
  k_s_init<<<(TBN + 255) / 256, blk, 0, stream>>>(obs, deg, s);
  long long nedge = (long long)Ec * Bc * Tc;  // 25.6M
  k_s_edges<<<(unsigned)((nedge + 255) / 256), blk, 0, stream>>>(ei, nrm, obs, s);
  k_copy_other<<<(Bc * Tc * Nc * (Fc - 1) + 255) / 256, blk, 0, stream>>>(obs, out);
  k_uext<<<(3 * 68 * 64 + 255) / 256, blk, 0, stream>>>(
      Wc_z, bc_z, Wl_z, bl_z, Wc_r, bc_r, Wl_r, bl_r, Wc_h, bc_h, Wl_h, bl_h, Uext);

  for (int t = 0; t < Tc; ++t) {
    const float* Hcur = (t & 1) ? Hb : Ha;
    float* Hnext      = (t & 1) ? Ha : Hb;
    k_gru_step<<<BN / 16 / 8, blk, 0, stream>>>(Hcur, Hnext, s + (size_t)t * BN, Uext, out, t);
  }
}